// AttnDecoder_17471926960376
// MI455X (gfx1250) — compile-verified
//
#include <hip/hip_runtime.h>

// ---------------------------------------------------------------------------
// MI455X (gfx1250) attention-decoder step. Memory-bound GEMV chain:
// ~560 MB fp32 weights @ 23.3 TB/s -> ~24 us roofline. All GEMVs run through
// V_WMMA_F32_16X16X4_F32: one wave32 produces 16 output rows, streaming the
// weight matrix with coalesced non-temporal b128 loads (one b128 feeds two
// WMMAs via a K-permutation that is applied identically to A and B).
// Log-softmax statistics are computed online inside the vocab GEMV so no
// kernel ever streams the 512 KB logit array through a single workgroup.
// ---------------------------------------------------------------------------

typedef __attribute__((ext_vector_type(2))) float v2f;
typedef __attribute__((ext_vector_type(4))) float v4f;
typedef __attribute__((ext_vector_type(8))) float v8f;

#define HID   1024
#define VOCAB 128000
#define LSTEP 100

// Output layout (floats): [0,128000) log_softmax, [128000,129024) h_new,
// [129024,129124) attn_weights.

// ---------------------------------------------------------------------------
// Kernel 1: embedding gather + attention logits + softmax + attn_applied.
// One block of 256 threads (8 waves). Writes attn_weights to d_out and
// cvec = concat(embedded, attn_applied) [2048] to ws.
// ---------------------------------------------------------------------------
__global__ void attn_kernel(const int* __restrict__ token,
                            const float* __restrict__ hidden,
                            const float* __restrict__ enc,
                            const float* __restrict__ emb,
                            const float* __restrict__ attn_W,
                            const float* __restrict__ attn_b,
                            float* __restrict__ d_out,
                            float* __restrict__ cvec) {
    __shared__ __align__(16) float semb[HID];
    __shared__ __align__(16) float sh[HID];
    __shared__ float slog[128];
    __shared__ float sw[128];
    __shared__ float sred[2];

    const int tid = threadIdx.x;               // 0..255
    const int tok = token[0];

    for (int j = tid; j < HID; j += 256) {
        float e = emb[(size_t)tok * HID + j];
        semb[j] = e;
        cvec[j] = e;                           // first half of combined vec
        sh[j]   = hidden[j];
    }
    __syncthreads();

    const int wv = tid >> 5, lane = tid & 31;

    // attn_logits[l] = attn_W[l,:] . concat(embedded, h) + attn_b[l]
    for (int l = wv; l < LSTEP; l += 8) {
        const float* wr = attn_W + (size_t)l * (2 * HID);
        float s = 0.0f;
        for (int j = lane; j < 2 * HID; j += 32) {
            float v = (j < HID) ? semb[j] : sh[j - HID];
            s += wr[j] * v;
        }
        for (int off = 16; off > 0; off >>= 1) s += __shfl_xor(s, off, 32);
        if (lane == 0) slog[l] = s + attn_b[l];
    }
    __syncthreads();

    // softmax over 100 logits (wave 0)
    if (wv == 0) {
        float m = -__builtin_inff();
        for (int l = lane; l < LSTEP; l += 32) m = fmaxf(m, slog[l]);
        for (int off = 16; off > 0; off >>= 1) m = fmaxf(m, __shfl_xor(m, off, 32));
        float ss = 0.0f;
        for (int l = lane; l < LSTEP; l += 32) ss += expf(slog[l] - m);
        for (int off = 16; off > 0; off >>= 1) ss += __shfl_xor(ss, off, 32);
        if (lane == 0) { sred[0] = m; sred[1] = ss; }
    }
    __syncthreads();

    const float mx = sred[0], inv = 1.0f / sred[1];
    if (tid < LSTEP) {
        float wl = expf(slog[tid] - mx) * inv;
        sw[tid] = wl;
        d_out[VOCAB + HID + tid] = wl;         // attn_weights output
    }
    __syncthreads();

    // attn_applied[j] = sum_l sw[l] * enc[l,j]  (coalesced over j)
    for (int j = tid; j < HID; j += 256) {
        float s = 0.0f;
        for (int l = 0; l < LSTEP; ++l) s += sw[l] * enc[(size_t)l * HID + j];
        cvec[HID + j] = s;                     // second half of combined vec
    }
}

// ---------------------------------------------------------------------------
// Generic WMMA f32 GEMV: out[r] = W[r,:] . vec + bias[r]  (optionally ReLU).
// W row-major [rows, K], K % 16 == 0. Block = 256 threads = 8 waves, each
// wave computes 16 rows; grid.x = rows/128.
//
// Per-lane b128 weight load covers k = k0+4t .. k0+4t+3 (t = lane>>4). Its
// low/high halves feed two WMMAs; A and B use the same K-permutation so each
// k contributes exactly once. If pmax/psum are non-null, an online softmax
// partial (block max, sum of exp) is emitted per block for log_softmax.
// ---------------------------------------------------------------------------
__global__ void gemv_wmma_kernel(const float* __restrict__ W,
                                 const float* __restrict__ vec,
                                 const float* __restrict__ bias,
                                 float* __restrict__ out,
                                 int K, int relu,
                                 float* __restrict__ pmax,
                                 float* __restrict__ psum) {
    __shared__ __align__(16) float svec[2048];
    __shared__ float spm[16];
    __shared__ float sps[16];
    const int tid = threadIdx.x;
    for (int j = tid * 4; j < K; j += 1024)
        *(v4f*)(svec + j) = *(const v4f*)(vec + j);
    __syncthreads();

    const int wv   = tid >> 5;
    const int lane = tid & 31;
    const int m    = lane & 15;       // output row within tile
    const int t    = lane >> 4;       // lane half
    const int row0 = blockIdx.x * 128 + wv * 16;

    const float* p  = W + (size_t)(row0 + m) * K + 4 * t;
    const float* pv = svec + 4 * t;

    v8f a0 = {}, a1 = {}, a2 = {}, a3 = {};
    for (int k0 = 0; k0 < K; k0 += 16) {
        v4f w0 = __builtin_nontemporal_load((const v4f*)(p + k0));
        v4f x0 = *(const v4f*)(pv + k0);
        v2f wA = __builtin_shufflevector(w0, w0, 0, 1);
        v2f xA = __builtin_shufflevector(x0, x0, 0, 1);
        a0 = __builtin_amdgcn_wmma_f32_16x16x4_f32(false, wA, false, xA,
                                                   (short)0, a0, false, false);
        v2f wB = __builtin_shufflevector(w0, w0, 2, 3);
        v2f xB = __builtin_shufflevector(x0, x0, 2, 3);
        a1 = __builtin_amdgcn_wmma_f32_16x16x4_f32(false, wB, false, xB,
                                                   (short)0, a1, false, false);
        v4f w1 = __builtin_nontemporal_load((const v4f*)(p + k0 + 8));
        v4f x1 = *(const v4f*)(pv + k0 + 8);
        v2f wC = __builtin_shufflevector(w1, w1, 0, 1);
        v2f xC = __builtin_shufflevector(x1, x1, 0, 1);
        a2 = __builtin_amdgcn_wmma_f32_16x16x4_f32(false, wC, false, xC,
                                                   (short)0, a2, false, false);
        v2f wD = __builtin_shufflevector(w1, w1, 2, 3);
        v2f xD = __builtin_shufflevector(x1, x1, 2, 3);
        a3 = __builtin_amdgcn_wmma_f32_16x16x4_f32(false, wD, false, xD,
                                                   (short)0, a3, false, false);
    }
    v8f acc = (a0 + a1) + (a2 + a3);

    // D layout: VGPR r -> row r (lanes 0-15) / row 8+r (lanes 16-31); all N
    // columns equal (B replicated). Lanes 0 and 16 each own 8 result rows.
    const bool writer = (lane & 15) == 0;
    float vals[8];
    float lm = -__builtin_inff();
    if (writer) {
        const int rbase = row0 + t * 8;
        #pragma unroll
        for (int r = 0; r < 8; ++r) {
            float v = acc[r] + bias[rbase + r];
            if (relu) v = fmaxf(v, 0.0f);
            out[rbase + r] = v;
            vals[r] = v;
            lm = fmaxf(lm, v);
        }
    }
    if (pmax != nullptr) {                     // uniform branch
        if (writer) {
            float ls = 0.0f;
            #pragma unroll
            for (int r = 0; r < 8; ++r) ls += expf(vals[r] - lm);
            spm[wv * 2 + t] = lm;
            sps[wv * 2 + t] = ls;
        }
        __syncthreads();
        if (tid == 0) {
            float M = -__builtin_inff();
            #pragma unroll
            for (int i = 0; i < 16; ++i) M = fmaxf(M, spm[i]);
            float S = 0.0f;
            #pragma unroll
            for (int i = 0; i < 16; ++i) S += sps[i] * expf(spm[i] - M);
            pmax[blockIdx.x] = M;
            psum[blockIdx.x] = S;
        }
    }
}

// ---------------------------------------------------------------------------
// GRU gate elementwise math (PyTorch order r,z,n). grid 4 x 256.
// ---------------------------------------------------------------------------
__global__ void gru_gate_kernel(const float* __restrict__ gi,
                                const float* __restrict__ gh,
                                const float* __restrict__ hidden,
                                float* __restrict__ d_out,
                                float* __restrict__ hnew) {
    const int i = blockIdx.x * 256 + threadIdx.x;
    float r = 1.0f / (1.0f + expf(-(gi[i]           + gh[i])));
    float z = 1.0f / (1.0f + expf(-(gi[HID + i]     + gh[HID + i])));
    float n = tanhf(gi[2 * HID + i] + r * gh[2 * HID + i]);
    float h = hidden[i];
    float o = (1.0f - z) * n + z * h;
    d_out[VOCAB + i] = o;                      // h_new output
    hnew[i] = o;
}

// ---------------------------------------------------------------------------
// Combine 1000 per-block online-softmax partials -> red[0]=max, red[1]=lse.
// Single block of 256, reads only 8 KB.
// ---------------------------------------------------------------------------
__global__ void lse_combine_kernel(const float* __restrict__ pmax,
                                   const float* __restrict__ psum,
                                   int n, float* __restrict__ red) {
    __shared__ float sm[256];
    __shared__ float ss[256];
    const int tid = threadIdx.x;
    float M = -__builtin_inff(), S = 0.0f;
    for (int i = tid; i < n; i += 256) {
        float m2 = pmax[i], s2 = psum[i];
        float Mn = fmaxf(M, m2);
        S = S * expf(M - Mn) + s2 * expf(m2 - Mn);
        M = Mn;
    }
    sm[tid] = M; ss[tid] = S;
    __syncthreads();
    for (int s = 128; s > 0; s >>= 1) {
        if (tid < s) {
            float M1 = sm[tid], S1 = ss[tid];
            float M2 = sm[tid + s], S2 = ss[tid + s];
            float Mn = fmaxf(M1, M2);
            sm[tid] = Mn;
            ss[tid] = S1 * expf(M1 - Mn) + S2 * expf(M2 - Mn);
        }
        __syncthreads();
    }
    if (tid == 0) { red[0] = sm[0]; red[1] = logf(ss[0]); }
}

__global__ void logsoftmax_kernel(const float* __restrict__ logits,
                                  const float* __restrict__ red,
                                  float* __restrict__ out) {
    const int v = blockIdx.x * 256 + threadIdx.x;  // grid = 500
    out[v] = logits[v] - red[0] - red[1];
}

// ---------------------------------------------------------------------------
extern "C" void kernel_launch(void* const* d_in, const int* in_sizes, int n_in,
                              void* d_out, int out_size, void* d_ws, size_t ws_size,
                              hipStream_t stream) {
    (void)in_sizes; (void)n_in; (void)out_size; (void)ws_size;
    const int*   token  = (const int*)  d_in[0];
    const float* hidden = (const float*)d_in[1];
    const float* enc    = (const float*)d_in[2];
    const float* emb    = (const float*)d_in[3];
    const float* attn_W = (const float*)d_in[4];
    const float* attn_b = (const float*)d_in[5];
    const float* comb_W = (const float*)d_in[6];
    const float* comb_b = (const float*)d_in[7];
    const float* W_ih   = (const float*)d_in[8];
    const float* W_hh   = (const float*)d_in[9];
    const float* b_ih   = (const float*)d_in[10];
    const float* b_hh   = (const float*)d_in[11];
    const float* out_W  = (const float*)d_in[12];
    const float* out_b  = (const float*)d_in[13];

    float* out    = (float*)d_out;
    float* ws     = (float*)d_ws;
    float* cvec   = ws;            // 2048   concat(embedded, attn_applied)
    float* xbuf   = ws + 2048;     // 1024   relu combine output
    float* gi     = ws + 3072;     // 3072
    float* gh     = ws + 6144;     // 3072
    float* hnew   = ws + 9216;     // 1024
    float* logits = ws + 10240;    // 128000
    float* pmax   = ws + 138240;   // 1000   per-block online-softmax max
    float* psum   = ws + 139240;   // 1000   per-block online-softmax sum
    float* red    = ws + 140240;   // 2      global max / log-sum-exp

    attn_kernel<<<1, 256, 0, stream>>>(token, hidden, enc, emb, attn_W, attn_b,
                                       out, cvec);
    // x = relu(comb_W @ cvec + comb_b)   [1024 rows, K=2048]
    gemv_wmma_kernel<<<8, 256, 0, stream>>>(comb_W, cvec, comb_b, xbuf,
                                            2048, 1, nullptr, nullptr);
    // gi = W_ih @ x + b_ih, gh = W_hh @ h + b_hh   [3072 rows, K=1024]
    gemv_wmma_kernel<<<24, 256, 0, stream>>>(W_ih, xbuf, b_ih, gi,
                                             1024, 0, nullptr, nullptr);
    gemv_wmma_kernel<<<24, 256, 0, stream>>>(W_hh, hidden, b_hh, gh,
                                             1024, 0, nullptr, nullptr);
    gru_gate_kernel<<<4, 256, 0, stream>>>(gi, gh, hidden, out, hnew);
    // logits = out_W @ h_new + out_b  [128000 rows, K=1024] — the 524 MB one,
    // with fused per-block online-softmax partials.
    gemv_wmma_kernel<<<1000, 256, 0, stream>>>(out_W, hnew, out_b, logits,
                                               1024, 0, pmax, psum);
    lse_combine_kernel<<<1, 256, 0, stream>>>(pmax, psum, 1000, red);
    logsoftmax_kernel<<<500, 256, 0, stream>>>(logits, red, out);
}